// ResGCNBlock_8881992368542
// MI455X (gfx1250) — compile-verified
//
#include <hip/hip_runtime.h>
#include <cstdint>
#include <cstddef>

// ---------------------------------------------------------------------------
// ResGCN block for MI455X (gfx1250, wave32).
//   h   = x @ W^T + b                      (f32 WMMA 16x16x4, 16x128 strip/wave)
//   deg = 1 + histogram(edge_index[0])     (atomic f32)
//   agg = scatter_add(norm * h[src] -> dst) + self-loop term
//   out = LayerNorm(ReLU(agg)) * gamma + beta + x
// ---------------------------------------------------------------------------

typedef float v2f __attribute__((ext_vector_type(2)));
typedef float v8f __attribute__((ext_vector_type(8)));

#define D 128
#define LN_EPS 1e-5f

// ---------------------------------------------------------------- init ------
__global__ void rgcn_init_kernel(float* __restrict__ deg,
                                 float* __restrict__ agg, int N) {
    int tid = blockIdx.x * blockDim.x + threadIdx.x;
    if (tid < N * D) agg[tid] = 0.0f;
    if (tid < N)     deg[tid] = 1.0f;   // self-loop contribution
}

// -------------------------------------------------------------- degree ------
__global__ void rgcn_degree_kernel(const int* __restrict__ row,
                                   float* __restrict__ deg, int E) {
    int e = blockIdx.x * blockDim.x + threadIdx.x;
    if (e < E) atomicAdd(&deg[row[e]], 1.0f);
}

// ---------------------------------------------------------------- gemm ------
// One wave -> one 16x128 output strip (16 rows, all 8 column tiles of D=128).
// Per K-step (K += 4): one A-fragment load feeds 8 independent
// V_WMMA_F32_16X16X4_F32 against 8 B-fragments -> x is read exactly once.
//
// A (16x4 f32) per-lane fragment, lane = 16*half + m:
//   a = { x[row0+m, k+2*half], x[row0+m, k+2*half+1] }
// B (4x16) with B[k][n] = W[n][k] (tile t covers cols 16t..16t+15):
//   b = { W[16t+m, k+2*half], W[16t+m, k+2*half+1] }
// C/D (16x16 f32): VGPR v, lane half h -> row = v + 8h, col = m.
__global__ void __launch_bounds__(256)
rgcn_gemm_kernel(const float* __restrict__ x, const float* __restrict__ W,
                 const float* __restrict__ bias, float* __restrict__ h, int N) {
    const int lane = threadIdx.x & 31;
    const int wave = threadIdx.x >> 5;
    const int m    = lane & 15;
    const int half = lane >> 4;
    const int row0 = blockIdx.x * 128 + wave * 16;   // 8 waves * 16 rows/block

    int r = row0 + m;
    if (r >= N) r = N - 1;              // clamp: keeps EXEC all-ones for WMMA

    const float* __restrict__ xrow  = x + (size_t)r * D + 2 * half;
    const float* __restrict__ wbase = W + (size_t)m * D + 2 * half;

    v8f acc[8] = {};
#pragma unroll 4
    for (int k = 0; k < D; k += 4) {
        v2f a = *(const v2f*)(xrow + k);
#pragma unroll
        for (int t = 0; t < 8; ++t) {
            v2f b = *(const v2f*)(wbase + (size_t)t * 16 * D + k);
            acc[t] = __builtin_amdgcn_wmma_f32_16x16x4_f32(
                /*neg_a=*/false, a, /*neg_b=*/false, b,
                /*c_mod=*/(short)0, acc[t], /*reuse_a=*/false, /*reuse_b=*/false);
        }
    }

#pragma unroll
    for (int t = 0; t < 8; ++t) {
        const float bb = bias[t * 16 + m];
#pragma unroll
        for (int v = 0; v < 8; ++v) {
            int orow = row0 + half * 8 + v;
            if (orow < N)
                h[(size_t)orow * D + t * 16 + m] = acc[t][v] + bb;
        }
    }
}

// ---------------------------------------------------------------- edges -----
// One wave per edge: lane d handles channels [4d, 4d+4). norm = rsqrt(ds*dt).
// h (25.6 MB) is fully L2-resident (192 MB L2), so the random gather and the
// f32 scatter atomics both run at L2 speed.
__global__ void __launch_bounds__(256)
rgcn_edge_kernel(const int* __restrict__ ei, const float* __restrict__ deg,
                 const float* __restrict__ h, float* __restrict__ agg, int E) {
    const int e = blockIdx.x * 8 + (threadIdx.x >> 5);
    if (e >= E) return;
    const int lane = threadIdx.x & 31;

    const int s = ei[e];        // source (row): message h[s]
    const int t = ei[E + e];    // target (col): aggregate here
    const float norm = __frsqrt_rn(deg[s] * deg[t]);

    const float4 v = ((const float4*)(h + (size_t)s * D))[lane];
    float* at = agg + (size_t)t * D + lane * 4;
    atomicAdd(at + 0, norm * v.x);
    atomicAdd(at + 1, norm * v.y);
    atomicAdd(at + 2, norm * v.z);
    atomicAdd(at + 3, norm * v.w);
}

// ------------------------------------------------- relu+LN+residual ---------
// One wave per node; 4 channels per lane; wave32 xor-shuffle reduction.
__global__ void __launch_bounds__(256)
rgcn_final_kernel(const float* __restrict__ x, const float* __restrict__ deg,
                  const float* __restrict__ h, const float* __restrict__ agg,
                  const float* __restrict__ gamma, const float* __restrict__ beta,
                  float* __restrict__ out, int N) {
    const int n = blockIdx.x * 8 + (threadIdx.x >> 5);
    if (n >= N) return;
    const int lane = threadIdx.x & 31;

    const float self_w = 1.0f / deg[n];   // dis[n]^2 for the self-loop edge

    const float4 hv = ((const float4*)(h   + (size_t)n * D))[lane];
    const float4 av = ((const float4*)(agg + (size_t)n * D))[lane];

    float4 v;
    v.x = fmaxf(av.x + self_w * hv.x, 0.0f);
    v.y = fmaxf(av.y + self_w * hv.y, 0.0f);
    v.z = fmaxf(av.z + self_w * hv.z, 0.0f);
    v.w = fmaxf(av.w + self_w * hv.w, 0.0f);

    float s  = v.x + v.y + v.z + v.w;
    float s2 = v.x * v.x + v.y * v.y + v.z * v.z + v.w * v.w;
#pragma unroll
    for (int o = 16; o >= 1; o >>= 1) {
        s  += __shfl_xor(s,  o, 32);
        s2 += __shfl_xor(s2, o, 32);
    }
    const float mu   = s * (1.0f / D);
    const float var  = s2 * (1.0f / D) - mu * mu;   // population var (ddof=0)
    const float rstd = __frsqrt_rn(var + LN_EPS);

    const float4 g  = ((const float4*)gamma)[lane];
    const float4 be = ((const float4*)beta)[lane];
    const float4 xv = ((const float4*)(x + (size_t)n * D))[lane];

    float4 r;
    r.x = (v.x - mu) * rstd * g.x + be.x + xv.x;
    r.y = (v.y - mu) * rstd * g.y + be.y + xv.y;
    r.z = (v.z - mu) * rstd * g.z + be.z + xv.z;
    r.w = (v.w - mu) * rstd * g.w + be.w + xv.w;

    ((float4*)(out + (size_t)n * D))[lane] = r;
}

// ---------------------------------------------------------------------------
extern "C" void kernel_launch(void* const* d_in, const int* in_sizes, int n_in,
                              void* d_out, int out_size, void* d_ws, size_t ws_size,
                              hipStream_t stream) {
    const float* x     = (const float*)d_in[0];
    const int*   ei    = (const int*)  d_in[1];   // [2, E] flat: row then col
    const float* W     = (const float*)d_in[2];
    const float* bias  = (const float*)d_in[3];
    const float* gamma = (const float*)d_in[4];
    const float* beta  = (const float*)d_in[5];

    const int N = in_sizes[0] / D;
    const int E = in_sizes[1] / 2;

    // workspace layout: deg[N] | h[N*D] | agg[N*D]   (~51.4 MB total)
    char* ws = (char*)d_ws;
    float* deg = (float*)ws;
    size_t off = ((size_t)N * sizeof(float) + 255) & ~(size_t)255;
    float* h = (float*)(ws + off);
    off += (size_t)N * D * sizeof(float);
    float* agg = (float*)(ws + off);

    rgcn_init_kernel<<<(N * D + 255) / 256, 256, 0, stream>>>(deg, agg, N);
    rgcn_degree_kernel<<<(E + 255) / 256, 256, 0, stream>>>(ei, deg, E);
    rgcn_gemm_kernel<<<(N + 127) / 128, 256, 0, stream>>>(x, W, bias, h, N);
    rgcn_edge_kernel<<<(E + 7) / 8, 256, 0, stream>>>(ei, deg, h, agg, E);
    rgcn_final_kernel<<<(N + 7) / 8, 256, 0, stream>>>(x, deg, h, agg, gamma,
                                                       beta, (float*)d_out, N);
}